// GCNLayer_27831388078276
// MI455X (gfx1250) — compile-verified
//
#include <hip/hip_runtime.h>
#include <hip/hip_bf16.h>

typedef __attribute__((ext_vector_type(2))) float v2f;
typedef __attribute__((ext_vector_type(8))) float v8f;

#define IN_DIM 128
#define OUT_DIM 128

// ---------------------------------------------------------------------------
// deg[i] = 1.0 (self loop)
__global__ void init_deg_kernel(float* __restrict__ deg, int n) {
    int i = blockIdx.x * blockDim.x + threadIdx.x;
    if (i < n) deg[i] = 1.0f;
}

// deg[dst[e]] += 1 for every edge
__global__ void deg_accum_kernel(const int* __restrict__ dst,
                                 float* __restrict__ deg, int e) {
    int i = blockIdx.x * blockDim.x + threadIdx.x;
    if (i < e) atomicAdd(&deg[dst[i]], 1.0f);
}

// ---------------------------------------------------------------------------
// h_ = h @ W^T + b  via v_wmma_f32_16x16x4_f32.
// Block = 8 waves; block b -> rows [16b,16b+16), wave w -> cols [16w,16w+16).
__global__ void __launch_bounds__(256)
gemm_wmma_kernel(const float* __restrict__ h,
                 const float* __restrict__ W,
                 const float* __restrict__ bias,
                 float* __restrict__ hw, int n) {
    const int wave = threadIdx.x >> 5;       // 0..7
    const int lane = threadIdx.x & 31;
    const int m0 = blockIdx.x * 16;
    const int n0 = wave * 16;

    const int laneMod  = lane & 15;          // M (for A) / N (for B)
    const int laneHalf = lane >> 4;          // selects K pair {0,1} vs {2,3}
    const int kOff     = laneHalf * 2;

    // Clamp row for loads so EXEC stays all-ones (WMMA requirement).
    int mrow = m0 + laneMod; if (mrow >= n) mrow = n - 1;
    const float* __restrict__ hrow = h + (size_t)mrow * IN_DIM;
    const float* __restrict__ wrow = W + (size_t)(n0 + laneMod) * IN_DIM;

    v8f acc = {};
#pragma unroll
    for (int k = 0; k < IN_DIM; k += 4) {
        v2f a, bm;
        a.x  = hrow[k + kOff];
        a.y  = hrow[k + kOff + 1];
        bm.x = wrow[k + kOff];
        bm.y = wrow[k + kOff + 1];
        // D = A(16x4) x B(4x16) + C ; emits v_wmma_f32_16x16x4_f32
        acc = __builtin_amdgcn_wmma_f32_16x16x4_f32(
            /*neg_a=*/false, a, /*neg_b=*/false, bm,
            /*c_mod=*/(short)0, acc, /*reuse_a=*/false, /*reuse_b=*/false);
    }

    const int col = n0 + laneMod;
    const float bn = bias[col];
#pragma unroll
    for (int j = 0; j < 8; ++j) {
        int row = m0 + j + laneHalf * 8;
        if (row < n) hw[(size_t)row * OUT_DIM + col] = acc[j] + bn;
    }
}

// ---------------------------------------------------------------------------
// agg[i][c] = h_[i][c] / deg[i]   (self-loop contribution, ew = 1/deg)
__global__ void self_init_kernel(const float* __restrict__ hw,
                                 const float* __restrict__ deg,
                                 float* __restrict__ out, int n) {
    int i = blockIdx.x * blockDim.x + threadIdx.x;
    int total = n * OUT_DIM;
    if (i < total) {
        int node = i >> 7;                    // /OUT_DIM
        out[i] = hw[i] * __frcp_rn(deg[node]);
    }
}

// One wave per edge: lane reads float4 of h_[src], 4 atomic f32 adds to agg[dst].
__global__ void edge_scatter_kernel(const int* __restrict__ src,
                                    const int* __restrict__ dst,
                                    const float* __restrict__ hw,
                                    const float* __restrict__ deg,
                                    float* __restrict__ out, int e) {
    int gid  = blockIdx.x * blockDim.x + threadIdx.x;
    int edge = gid >> 5;
    int lane = gid & 31;
    if (edge >= e) return;
    int s = src[edge];
    int d = dst[edge];
    float ew = __frsqrt_rn(deg[s] * deg[d]);
    const float4* __restrict__ hs = (const float4*)(hw + (size_t)s * OUT_DIM);
    float4 v = hs[lane];
    float* o = out + (size_t)d * OUT_DIM + lane * 4;
    atomicAdd(o + 0, v.x * ew);
    atomicAdd(o + 1, v.y * ew);
    atomicAdd(o + 2, v.z * ew);
    atomicAdd(o + 3, v.w * ew);
}

// ---------------------------------------------------------------------------
extern "C" void kernel_launch(void* const* d_in, const int* in_sizes, int n_in,
                              void* d_out, int out_size, void* d_ws, size_t ws_size,
                              hipStream_t stream) {
    const float* h    = (const float*)d_in[0];   // [N, 128]
    const float* W    = (const float*)d_in[1];   // [128, 128]
    const float* bias = (const float*)d_in[2];   // [128]
    const int*   edges = (const int*)d_in[3];    // [2, E]

    const int N = in_sizes[0] / IN_DIM;
    const int E = in_sizes[3] / 2;
    const int* src = edges;
    const int* dst = edges + E;

    float* hw  = (float*)d_ws;                   // [N,128] transformed features
    float* deg = hw + (size_t)N * OUT_DIM;       // [N] degrees

    float* out = (float*)d_out;                  // [N,128] aggregated output

    // 1) degrees (self loop = 1, plus in-degree over edges)
    init_deg_kernel<<<(N + 255) / 256, 256, 0, stream>>>(deg, N);
    deg_accum_kernel<<<(E + 255) / 256, 256, 0, stream>>>(dst, deg, E);

    // 2) linear transform via FP32 WMMA
    gemm_wmma_kernel<<<(N + 15) / 16, 256, 0, stream>>>(h, W, bias, hw, N);

    // 3) self-loop init of output, then edge-weighted scatter-add
    int total = N * OUT_DIM;
    self_init_kernel<<<(total + 255) / 256, 256, 0, stream>>>(hw, deg, out, N);

    long long threads = (long long)E * 32;
    edge_scatter_kernel<<<(int)((threads + 255) / 256), 256, 0, stream>>>(
        src, dst, hw, deg, out, E);
}